// BiMamba2DSlow4_31971736552199
// MI455X (gfx1250) — compile-verified
//
#include <hip/hip_runtime.h>
#include <hip/hip_bf16.h>
#include <math.h>

// ---------------------------------------------------------------------------
// BiMamba2D (VMamba SS2D) for MI455X / gfx1250, wave32, fp32 WMMA 16x16x4.
//
// Pipeline:
//  k_prep     : transpose in_proj/out_proj weights; build Wcomb[k] (224x192) =
//               [dtw@xpw[0:6] ; xpw[6:22] ; xpw[22:38]]  (dt_proj folded in)
//  k_inproj   : WMMA GEMM  [8192x96]x[96x384] -> xi (NCHW) + z ([b,l,192])
//  k_conv     : depthwise 3x3 + bias + SiLU -> xc (HW order) and xc_wh (WH order)
//  k_projgemm : per-direction WMMA GEMM [8192x192]x[192x224] -> proj[k][b][224][l]
//               rows 0..191 = delta_raw (pre-softplus), 192..207 = B, 208..223 = C
//  k_scan     : 1536 scans (k,b,d); 16 lanes = 16 states; forward/backward per k;
//               writes y4 in natural HW position (handles flip/transpose mapping);
//               explicit global_prefetch on the 4 streaming inputs
//  k_combine  : sum 4 dirs + LayerNorm(192) + silu(z) gate -> g [8192x192]
//  k_outproj  : WMMA GEMM [8192x192]x[192x96] -> d_out
// ---------------------------------------------------------------------------

typedef __attribute__((ext_vector_type(2))) float v2f;
typedef __attribute__((ext_vector_type(8))) float v8f;

#define Bb 2
#define Hh 64
#define Ww 64
#define LL 4096          // H*W
#define DM 96            // d_model
#define DI 192           // d_inner
#define DS 16            // d_state
#define DR 6             // dt_rank
#define KK 4             // directions
#define NPRJ 224         // 192 delta + 16 B + 16 C
#define MROWS 8192       // B*L

// ---- generic fp32 16x16 WMMA tile: C = A[16xK] * B[KxN-tile] -------------
// A element (row, c) at Abase[row*aRS + c*aCS]; B row-major [K][ldb].
__device__ __forceinline__ v8f wmma_tile_f32(const float* __restrict__ Abase,
                                             int aRS, int aCS,
                                             const float* __restrict__ Bbase,
                                             int ldb, int Kdim, int lane) {
  v8f acc;
  #pragma unroll
  for (int i = 0; i < 8; ++i) acc[i] = 0.f;
  const int row = lane & 15;          // A row / C col index
  const int col = lane & 15;          // B col
  const int kb  = (lane >> 4) << 1;   // K sub-offset per lane half
  for (int kk = 0; kk < Kdim; kk += 4) {
    v2f a, b;
    a.x = Abase[(size_t)row * aRS + (size_t)(kk + kb)     * aCS];
    a.y = Abase[(size_t)row * aRS + (size_t)(kk + kb + 1) * aCS];
    b.x = Bbase[(size_t)(kk + kb)     * ldb + col];
    b.y = Bbase[(size_t)(kk + kb + 1) * ldb + col];
    acc = __builtin_amdgcn_wmma_f32_16x16x4_f32(false, a, false, b,
                                                (short)0, acc, false, false);
  }
  return acc;
}

// ---- prep: weight transposes + combined projection matrix ------------------
__global__ void k_prep(const float* __restrict__ in_proj_w,   // [384][96]
                       const float* __restrict__ out_proj_w,  // [96][192]
                       const float* __restrict__ xpw,         // [4][38][192]
                       const float* __restrict__ dtw,         // [4][192][6]
                       float* __restrict__ Wt_in,             // [96][384]
                       float* __restrict__ Wt_out,            // [192][96]
                       float* __restrict__ Wcomb)             // [4][192][224]
{
  int gid = blockIdx.x * blockDim.x + threadIdx.x;
  const int N0 = DM * 2 * DI;            // 36864
  const int N1 = DI * DM;                // 18432
  const int N2 = KK * DI * NPRJ;         // 172032
  if (gid < N0) {
    int c = gid / (2 * DI), n = gid % (2 * DI);
    Wt_in[c * (2 * DI) + n] = in_proj_w[n * DM + c];
  } else if (gid < N0 + N1) {
    int i = gid - N0;
    int c = i / DM, n = i % DM;
    Wt_out[c * DM + n] = out_proj_w[n * DI + c];
  } else if (gid < N0 + N1 + N2) {
    int i = gid - N0 - N1;
    int k = i / (DI * NPRJ);
    int r = i % (DI * NPRJ);
    int c = r / NPRJ, np = r % NPRJ;
    float v;
    if (np < DI) {                        // delta_raw row: Weff = dtw @ xpw[0:6]
      v = 0.f;
      #pragma unroll
      for (int r6 = 0; r6 < DR; ++r6)
        v += dtw[(k * DI + np) * DR + r6] * xpw[(k * 38 + r6) * DI + c];
    } else if (np < DI + DS) {            // B rows 6..21
      v = xpw[(k * 38 + DR + (np - DI)) * DI + c];
    } else {                              // C rows 22..37
      v = xpw[(k * 38 + DR + DS + (np - DI - DS)) * DI + c];
    }
    Wcomb[((size_t)k * DI + c) * NPRJ + np] = v;
  }
}

// ---- in_proj GEMM: xz = x @ Wt_in ; split into xi (NCHW) and z -------------
__global__ void k_inproj(const float* __restrict__ x,      // [8192][96]
                         const float* __restrict__ Wt_in,  // [96][384]
                         float* __restrict__ xi,           // [b][192][4096]
                         float* __restrict__ z)            // [8192][192]
{
  int wave = (blockIdx.x * blockDim.x + threadIdx.x) >> 5;
  int lane = threadIdx.x & 31;
  const int nT = (2 * DI) / 16;          // 24
  if (wave >= (MROWS / 16) * nT) return;
  int tm = wave / nT, tn = wave % nT;
  int m0 = tm * 16, n0 = tn * 16;
  v8f acc = wmma_tile_f32(x + (size_t)m0 * DM, DM, 1, Wt_in + n0, 2 * DI, DM, lane);
  int half = lane >> 4, col = lane & 15;
  // b and the xi-vs-z split are wave-uniform (tiles are 16-aligned, 4096%16==0)
  int b  = m0 >> 12;
  int l0 = (m0 & (LL - 1)) + half * 8;
  if (n0 < DI) {
    float* dst = xi + ((size_t)b * DI + (n0 + col)) * LL + l0;
    #pragma unroll
    for (int i = 0; i < 8; ++i) dst[i] = acc[i];
  } else {
    float* dst = z + (size_t)(m0 + half * 8) * DI + (n0 - DI + col);
    #pragma unroll
    for (int i = 0; i < 8; ++i) dst[(size_t)i * DI] = acc[i];
  }
}

// ---- depthwise 3x3 conv + bias + SiLU; emit HW and WH orderings ------------
__global__ void k_conv(const float* __restrict__ xi, const float* __restrict__ cw,
                       const float* __restrict__ cb,
                       float* __restrict__ xc, float* __restrict__ xc_wh)
{
  int gid = blockIdx.x * blockDim.x + threadIdx.x;     // b*DI*LL
  if (gid >= Bb * DI * LL) return;
  int l = gid & (LL - 1);
  int d = (gid >> 12) % DI;
  int h = l >> 6, w = l & 63;
  const float* base = xi + ((size_t)(gid >> 12)) * LL;
  float s = cb[d];
  #pragma unroll
  for (int i = -1; i <= 1; ++i) {
    int hh = h + i;
    if (hh < 0 || hh >= Hh) continue;
    #pragma unroll
    for (int j = -1; j <= 1; ++j) {
      int ww2 = w + j;
      if (ww2 < 0 || ww2 >= Ww) continue;
      s += cw[d * 9 + (i + 1) * 3 + (j + 1)] * base[hh * Ww + ww2];
    }
  }
  float v = s / (1.f + __expf(-s));      // SiLU
  xc[gid] = v;
  xc_wh[((size_t)(gid >> 12)) * LL + (w * Hh + h)] = v;
}

// ---- per-direction projection GEMM: proj[k][b][n'][l] ----------------------
__global__ void k_projgemm(const float* __restrict__ xc, const float* __restrict__ xc_wh,
                           const float* __restrict__ Wcomb, float* __restrict__ proj)
{
  int wave = (blockIdx.x * blockDim.x + threadIdx.x) >> 5;
  int lane = threadIdx.x & 31;
  const int mT = MROWS / 16, nT = NPRJ / 16;           // 512, 14
  if (wave >= KK * mT * nT) return;
  int k = wave / (mT * nT);
  int r = wave % (mT * nT);
  int tm = r / nT, tn = r % nT;
  int m0 = tm * 16, n0 = tn * 16;
  int b = m0 >> 12, l0 = m0 & (LL - 1);
  const float* src = (k & 1) ? xc_wh : xc;
  // A element (row, c) = src[b][c][l0+row]  -> rowStride 1, colStride LL
  const float* Ab = src + (size_t)b * DI * LL + l0;
  v8f acc = wmma_tile_f32(Ab, 1, LL, Wcomb + (size_t)k * DI * NPRJ + n0, NPRJ, DI, lane);
  int half = lane >> 4, col = lane & 15;
  float* dst = proj + ((size_t)((k * Bb + b) * NPRJ) + (n0 + col)) * LL
                    + l0 + half * 8;
  #pragma unroll
  for (int i = 0; i < 8; ++i) dst[i] = acc[i];
}

// ---- selective scan: 2 scans per wave, 16 lanes = 16 states ----------------
#define PF_DIST 1024     // prefetch distance in elements (4 KB per stream)

__global__ void k_scan(const float* __restrict__ proj,
                       const float* __restrict__ xc, const float* __restrict__ xc_wh,
                       const float* __restrict__ A_logs,   // [4][192][16]
                       const float* __restrict__ Ds,       // [4][192]
                       const float* __restrict__ dt_bias,  // [4][192]
                       float* __restrict__ y4)             // [4][b][192][L] (HW pos)
{
  int lane = threadIdx.x & 31;
  int half = lane >> 4;
  int n = lane & 15;
  int s = blockIdx.x * 2 + half;                 // 0..1535
  // k, b, direction are wave-uniform: (2B, 2B+1) can't straddle a k/b boundary
  int k = s / (Bb * DI);
  int b = (s / DI) % Bb;
  int d = s % DI;
  const float a    = -__expf(A_logs[((size_t)k * DI + d) * DS + n]);
  const float bias = dt_bias[k * DI + d];
  const float Dk   = Ds[k * DI + d];
  const size_t pbase = (size_t)(k * Bb + b) * NPRJ * LL;
  const float* dptr = proj + pbase + (size_t)d * LL;
  const float* Bp   = proj + pbase + (size_t)(DI + n) * LL;
  const float* Cp   = proj + pbase + (size_t)(DI + DS + n) * LL;
  const float* up   = ((k & 1) ? xc_wh : xc) + ((size_t)b * DI + d) * LL;
  float* yp = y4 + ((size_t)(k * Bb + b) * DI + d) * LL;
  const bool rev = (k >= 2), whm = (k & 1);
  float h = 0.f;
  for (int t = 0; t < LL; ++t) {
    int l = rev ? (LL - 1 - t) : t;
    if ((t & 63) == 0 && (t + PF_DIST) < LL) {        // wave-uniform
      int lp = rev ? (LL - 1 - (t + PF_DIST)) : (t + PF_DIST);
      __builtin_prefetch(dptr + lp, 0, 1);
      __builtin_prefetch(Bp + lp, 0, 1);
      __builtin_prefetch(Cp + lp, 0, 1);
      __builtin_prefetch(up + lp, 0, 1);
    }
    float dr = dptr[l] + bias;
    float delta = (dr > 20.f) ? dr : log1pf(__expf(dr));   // softplus
    float u = up[l];
    float dA = __expf(delta * a);
    h = dA * h + (delta * u) * Bp[l];
    float yv = h * Cp[l];
    yv += __shfl_xor(yv, 8, 32);
    yv += __shfl_xor(yv, 4, 32);
    yv += __shfl_xor(yv, 2, 32);
    yv += __shfl_xor(yv, 1, 32);
    if (n == 0) {
      int pos = whm ? (((l & 63) << 6) | (l >> 6)) : l;    // WH -> HW position
      yp[pos] = yv + u * Dk;
    }
  }
}

// ---- combine 4 dirs + LayerNorm(192) + SiLU(z) gate ------------------------
__global__ void k_combine(const float* __restrict__ y4, const float* __restrict__ z,
                          const float* __restrict__ gam, const float* __restrict__ bet,
                          float* __restrict__ g)
{
  __shared__ float sp0[6], sp1[6], mv[2];
  int bl = blockIdx.x;                 // 0..8191  (b*L + l)
  int d = threadIdx.x;                 // 0..191
  int b = bl >> 12, l = bl & (LL - 1);
  float v = 0.f;
  #pragma unroll
  for (int k = 0; k < KK; ++k)
    v += y4[((size_t)((k * Bb + b) * DI) + d) * LL + l];
  float s1 = v, s2 = v * v;
  #pragma unroll
  for (int m = 16; m >= 1; m >>= 1) {
    s1 += __shfl_xor(s1, m, 32);
    s2 += __shfl_xor(s2, m, 32);
  }
  int wid = d >> 5;
  if ((d & 31) == 0) { sp0[wid] = s1; sp1[wid] = s2; }
  __syncthreads();
  if (d == 0) {
    float a = 0.f, c = 0.f;
    #pragma unroll
    for (int i = 0; i < 6; ++i) { a += sp0[i]; c += sp1[i]; }
    mv[0] = a / (float)DI; mv[1] = c / (float)DI;
  }
  __syncthreads();
  float mu = mv[0];
  float var = mv[1] - mu * mu;
  float yn = (v - mu) * rsqrtf(var + 1e-5f) * gam[d] + bet[d];
  float zv = z[(size_t)bl * DI + d];
  g[(size_t)bl * DI + d] = yn * (zv / (1.f + __expf(-zv)));
}

// ---- out_proj GEMM: d_out = g @ Wt_out -------------------------------------
__global__ void k_outproj(const float* __restrict__ g, const float* __restrict__ Wt_out,
                          float* __restrict__ out)
{
  int wave = (blockIdx.x * blockDim.x + threadIdx.x) >> 5;
  int lane = threadIdx.x & 31;
  const int nT = DM / 16;                      // 6
  if (wave >= (MROWS / 16) * nT) return;
  int tm = wave / nT, tn = wave % nT;
  int m0 = tm * 16, n0 = tn * 16;
  v8f acc = wmma_tile_f32(g + (size_t)m0 * DI, DI, 1, Wt_out + n0, DM, DI, lane);
  int half = lane >> 4, col = lane & 15;
  float* dst = out + (size_t)(m0 + half * 8) * DM + n0 + col;
  #pragma unroll
  for (int i = 0; i < 8; ++i) dst[(size_t)i * DM] = acc[i];
}

// ---------------------------------------------------------------------------
extern "C" void kernel_launch(void* const* d_in, const int* in_sizes, int n_in,
                              void* d_out, int out_size, void* d_ws, size_t ws_size,
                              hipStream_t stream) {
  const float* x        = (const float*)d_in[0];
  const float* in_w     = (const float*)d_in[1];
  const float* conv_w   = (const float*)d_in[2];
  const float* conv_b   = (const float*)d_in[3];
  const float* xpw      = (const float*)d_in[4];
  const float* dtw      = (const float*)d_in[5];
  const float* dt_bias  = (const float*)d_in[6];
  const float* A_logs   = (const float*)d_in[7];
  const float* Ds       = (const float*)d_in[8];
  const float* ln_g     = (const float*)d_in[9];
  const float* ln_b     = (const float*)d_in[10];
  const float* out_w    = (const float*)d_in[11];
  float* out = (float*)d_out;

  float* ws = (float*)d_ws;
  float* Wt_in  = ws;                    // 36864
  float* Wt_out = Wt_in  + 36864;        // 18432
  float* Wcomb  = Wt_out + 18432;        // 172032
  float* xi     = Wcomb  + 172032;       // 1572864
  float* zbuf   = xi     + 1572864;      // 1572864
  float* xc     = zbuf   + 1572864;      // 1572864
  float* xc_wh  = xc     + 1572864;      // 1572864
  float* proj   = xc_wh  + 1572864;      // 7340032
  float* y4     = proj   + 7340032;      // 6291456
  float* g      = y4     + 6291456;      // 1572864

  // prep: 227328 elements
  k_prep<<<888, 256, 0, stream>>>(in_w, out_w, xpw, dtw, Wt_in, Wt_out, Wcomb);
  // in_proj: 12288 tiles, 8 waves/block
  k_inproj<<<1536, 256, 0, stream>>>(x, Wt_in, xi, zbuf);
  // conv: 1572864 threads
  k_conv<<<6144, 256, 0, stream>>>(xi, conv_w, conv_b, xc, xc_wh);
  // projection GEMM: 4*512*14 = 28672 tiles
  k_projgemm<<<3584, 256, 0, stream>>>(xc, xc_wh, Wcomb, proj);
  // selective scan: 1536 scans, 2/wave
  k_scan<<<768, 32, 0, stream>>>(proj, xc, xc_wh, A_logs, Ds, dt_bias, y4);
  // combine + LN + gate
  k_combine<<<8192, 192, 0, stream>>>(y4, zbuf, ln_g, ln_b, g);
  // out_proj: 3072 tiles
  k_outproj<<<384, 256, 0, stream>>>(g, Wt_out, out);
}